// SE3Transformer_Vector_68813966016599
// MI455X (gfx1250) — compile-verified
//
#include <hip/hip_runtime.h>
#include <math.h>

// ---------------------------------------------------------------------------
// SE(3)-Transformer forward for MI455X (gfx1250, wave32, WMMA + async-LDS).
//
// Per 16-edge tile (one wave): radial MLP 5->32->32 via V_WMMA_F32_16X16X4_F32,
// then the W3 GEMM (N = flattened (o,c,f), K=32) in 16-wide chunks, each chunk
// folded immediately into t[e,o,p] against the basis-contracted y — the huge
// per-edge kernel tensor R never touches HBM.  Basis/x staging and the W3
// chunk stream use GLOBAL_LOAD_ASYNC_TO_LDS_B128 (ASYNCcnt), overlapped with
// the WMMA phases; single-wave blocks sync LDS with bare s_wait_dscnt.
// Final conv scatter-adds straight into node accumulators.
// ---------------------------------------------------------------------------

#define NE 131072
#define NN 16384

typedef __attribute__((ext_vector_type(2))) float v2f;
typedef __attribute__((ext_vector_type(8))) float v8f;

// single-wave workgroup: LDS producer->consumer only needs DScnt==0
__device__ inline void wave_lds_fence() {
  asm volatile("s_wait_dscnt 0x0" ::: "memory");
}
__device__ inline void wait_async_0() {
  asm volatile("s_wait_asynccnt 0x0" ::: "memory");
}
__device__ inline void wait_async_le4() {
  asm volatile("s_wait_asynccnt 0x4" ::: "memory");
}
// per-lane 16B copy global -> LDS, tracked by ASYNCcnt (no VGPR round-trip)
__device__ inline void async_b128_to_lds(void* ldsp, const void* gp) {
  unsigned lo = (unsigned)(size_t)ldsp;   // low 32 bits = LDS byte offset
  asm volatile("global_load_async_to_lds_b128 %0, %1, off"
               :: "v"(lo), "v"(gp) : "memory");
}

// ----------------------------- LDS layout ----------------------------------
template<int NCI,int NCO,int MI,int MO,int F>
struct PL {
  static constexpr int YFULL  = 16*NCI*MO*F;
  static constexpr int CSPLIT = (YFULL > 8192) ? 2 : 1;   // keep <=64KB LDS
  static constexpr int CB     = NCI / CSPLIT;
  static constexpr int SSRC = 0;                    // 16 ints (src gather)
  static constexpr int W1po = 16;                   // W1 padded [32][8]
  static constexpr int W2o  = W1po + 32*8;          // W2 [32][32]
  static constexpr int H2o  = W2o + 32*32;          // h2 [16][32]
  static constexpr int Xo   = H2o + 16*32;          // gathered x [16,NCI,MI]
  static constexpr int Bo   = Xo + 16*NCI*MI;       // basis tile [16,MO,MI,F]
  static constexpr int Yo   = Bo + 16*MO*MI*F;      // y part [16,CB,MO,F]
  static constexpr int SCR  = Yo + 16*CB*MO*F;      // scratch union:
  static constexpr int RWo  = SCR;                  //   MLP: rw [16,8]
  static constexpr int H1o  = SCR + 16*8;           //   MLP: h1 [16,32]
  static constexpr int Ro   = SCR;                  //   GEMM: R chunk [16,16]
  static constexpr int W3co = SCR + 16*16;          //   GEMM: W3 ping-pong [2][16][32]
  static constexpr int To   = SCR + 16*16 + (CSPLIT==1 ? 2*16*32 : 0);
  static constexpr int MLPEND = 16*8 + 16*32;
  static constexpr int TEND   = 16*16 + (CSPLIT==1 ? 2*16*32 : 0) + 16*NCO*MO;
  static constexpr int TOTAL  = SCR + (MLPEND > TEND ? MLPEND : TEND);
  static_assert(TOTAL <= 16384, "LDS budget (64KB) exceeded");
};

// ------------------------ fused pconv pair kernel --------------------------
// one wave (32 threads) per 16-edge tile
template<int NCI,int NCO,int MI,int MO,int F>
__global__ __launch_bounds__(32)
void pconv_kernel(const float* __restrict__ rp, const float* __restrict__ wp,
                  const int* __restrict__ src,
                  const float* __restrict__ W1, const float* __restrict__ b1,
                  const float* __restrict__ W2, const float* __restrict__ b2,
                  const float* __restrict__ W3, const float* __restrict__ b3,
                  const float* __restrict__ basis,
                  const float* __restrict__ x,
                  const int* __restrict__ sdst,      // non-null => scatter-add
                  float* __restrict__ out, int accumulate)
{
  using L = PL<NCI,NCO,MI,MO,F>;
  constexpr int S = MO*MI*F;                 // basis floats per edge
  extern __shared__ float smem[];
  const int lane = (int)threadIdx.x;
  const int e0   = (int)blockIdx.x * 16;
  const int lx   = lane & 15;     // edge-in-tile (A,D rows) / N column
  const int hi   = lane >> 4;     // K-pair half for A/B packing

  // prefetch next tile's basis (global_prefetch_b8)
  if ((int)blockIdx.x + 1 < (int)gridDim.x)
    __builtin_prefetch(&basis[(size_t)(e0 + 16) * S], 0, 1);

  int* sSrc = (int*)&smem[L::SSRC];
  if (lane < 16) sSrc[lane] = src[e0 + lane];

  // ---- kick off async basis tile staging (overlaps with whole MLP) -------
  {
    const float4* bv = (const float4*)(basis + (size_t)e0 * S);
    float4* bl = (float4*)&smem[L::Bo];
    for (int i = lane; i < (16*S)/4; i += 32)
      async_b128_to_lds(&bl[i], &bv[i]);
  }

  // ---- stage weights: W1 zero-padded to [32][8], W2, rw=[r,w,0] ----------
  for (int i = lane; i < 32*8; i += 32) {
    int n = i >> 3, k = i & 7;
    float v = W1[n*5 + (k < 5 ? k : 4)];          // clamped, unconditional
    smem[L::W1po + i] = (k < 5) ? v : 0.f;
  }
  for (int i = lane; i < (32*32)/4; i += 32)
    ((float4*)&smem[L::W2o])[i] = ((const float4*)W2)[i];
  if (lane < 16) smem[L::RWo + lane*8 + 0] = rp[e0 + lane];
  for (int i = lane; i < 64; i += 32) {           // 16 edges x 4 w-features
    int e = i >> 2, j = i & 3;
    smem[L::RWo + e*8 + 1 + j] = wp[(e0 + e)*4 + j];
  }
  for (int i = lane; i < 48; i += 32) {           // zero pad cols 5..7
    int e = i / 3, k = 5 + i % 3;
    smem[L::RWo + e*8 + k] = 0.f;
  }
  wave_lds_fence();

  // ---- gather x[src]: async b128 when layout allows, overlaps MLP too ----
  if constexpr ((NCI*MI) % 4 == 0) {
    constexpr int XV = (NCI*MI) / 4;
    float4* xl = (float4*)&smem[L::Xo];
    for (int i = lane; i < 16*XV; i += 32) {
      int e = i / XV;
      async_b128_to_lds(&xl[i], &((const float4*)x)[(size_t)sSrc[e]*XV + (i % XV)]);
    }
  } else {
    for (int i = lane; i < 16*NCI*MI; i += 32) {
      int e = i / (NCI*MI);
      smem[L::Xo + i] = x[(size_t)sSrc[e]*(NCI*MI) + (i % (NCI*MI))];
    }
  }

  // ---- h1 = relu(rw @ W1^T + b1), 16x32, K=5 padded to 8 ------------------
#pragma unroll
  for (int nc = 0; nc < 2; ++nc) {
    v8f acc = {};
#pragma unroll
    for (int kc = 0; kc < 2; ++kc) {
      int k = kc*4 + hi*2;
      v2f a, b;
      a.x = smem[L::RWo + lx*8 + k];
      a.y = smem[L::RWo + lx*8 + k + 1];
      int n = nc*16 + lx;
      b.x = smem[L::W1po + n*8 + k];
      b.y = smem[L::W1po + n*8 + k + 1];
      acc = __builtin_amdgcn_wmma_f32_16x16x4_f32(false, a, false, b, (short)0, acc, false, false);
    }
    int n = nc*16 + lx;
    float bb = b1[n];
#pragma unroll
    for (int rr = 0; rr < 8; ++rr) {
      float v = acc[rr] + bb;
      smem[L::H1o + (hi*8 + rr)*32 + n] = v > 0.f ? v : 0.f;
    }
  }
  wave_lds_fence();

  // ---- h2 = relu(h1 @ W2^T + b2) ------------------------------------------
#pragma unroll
  for (int nc = 0; nc < 2; ++nc) {
    v8f acc = {};
#pragma unroll
    for (int kc = 0; kc < 8; ++kc) {
      int k = kc*4 + hi*2;
      v2f a, b;
      a.x = smem[L::H1o + lx*32 + k];
      a.y = smem[L::H1o + lx*32 + k + 1];
      int n = nc*16 + lx;
      b.x = smem[L::W2o + n*32 + k];
      b.y = smem[L::W2o + n*32 + k + 1];
      acc = __builtin_amdgcn_wmma_f32_16x16x4_f32(false, a, false, b, (short)0, acc, false, false);
    }
    int n = nc*16 + lx;
    float bb = b2[n];
#pragma unroll
    for (int rr = 0; rr < 8; ++rr) {
      float v = acc[rr] + bb;
      smem[L::H2o + (hi*8 + rr)*32 + n] = v > 0.f ? v : 0.f;
    }
  }
  for (int i = lane; i < 16*NCO*MO; i += 32) smem[L::To + i] = 0.f;
  wave_lds_fence();

  // A operand of the W3 GEMM is h2: constant across all chunks -> hoist.
  v2f aw[8];
#pragma unroll
  for (int kc = 0; kc < 8; ++kc) {
    int k = kc*4 + hi*2;
    aw[kc].x = smem[L::H2o + lx*32 + k];
    aw[kc].y = smem[L::H2o + lx*32 + k + 1];
  }
  wait_async_0();             // basis tile + gathered x now resident in LDS

  // ---- c-split loop: y = basis . x, then W3 GEMM + fold into t ------------
  for (int cs = 0; cs < L::CSPLIT; ++cs) {
    const int c0 = cs * L::CB;
    // y[e,cl,p,f] = sum_q basis[e,p,q,f] * x[e,c0+cl,q]
    for (int i = lane; i < 16*L::CB*MO*F; i += 32) {
      int e   = i / (L::CB*MO*F);
      int rem = i % (L::CB*MO*F);
      int cl  = rem / (MO*F);
      int pf  = rem % (MO*F);
      int p = pf / F, f = pf % F;
      const float* bp = &smem[L::Bo + ((e*MO + p)*MI)*F + f];
      const float* xp = &smem[L::Xo + (e*NCI + c0 + cl)*MI];
      float acc = 0.f;
#pragma unroll
      for (int q = 0; q < MI; ++q) acc += bp[q*F] * xp[q];
      smem[L::Yo + i] = acc;
    }
    wave_lds_fence();

    constexpr int LTOT   = NCO * L::CB * F;
    constexpr int NCHUNK = (LTOT + 15) / 16;

    // async ping-pong stream of W3 chunk rows (contiguous when CSPLIT==1)
    auto issueW3 = [&](int lcc, int bufi) {
      const float4* W3v = (const float4*)W3;
      float4* dstl = (float4*)&smem[L::W3co + bufi*512];
#pragma unroll
      for (int j = 0; j < 4; ++j) {
        int i = lane + 32*j;                       // 128 float4 per chunk
        int row = i >> 3, kq = i & 7;
        int ngr = lcc*16 + row;
        int ngc = ngr < LTOT ? ngr : 0;            // clamp; junk cols unused
        async_b128_to_lds(&dstl[i], &W3v[(size_t)ngc*8 + kq]);
      }
    };
    if constexpr (L::CSPLIT == 1) issueW3(0, 0);

#pragma unroll 1
    for (int lc = 0; lc < NCHUNK; ++lc) {
      v8f acc = {};
      if constexpr (L::CSPLIT == 1) {
        if (lc + 1 < NCHUNK) { issueW3(lc + 1, (lc + 1) & 1); wait_async_le4(); }
        else                 { wait_async_0(); }
        const int bufo = L::W3co + (lc & 1)*512;
#pragma unroll
        for (int kc = 0; kc < 8; ++kc) {
          int k = kc*4 + hi*2;
          v2f b;
          b.x = smem[bufo + lx*32 + k];
          b.y = smem[bufo + lx*32 + k + 1];
          acc = __builtin_amdgcn_wmma_f32_16x16x4_f32(false, aw[kc], false, b, (short)0, acc, false, false);
        }
      } else {
        // non-contiguous rows: clamped loads (junk columns are never folded)
        int l   = lc*16 + lx;
        int o   = l / (L::CB * F);
        int clf = l % (L::CB * F);
        int cl = clf / F, f = clf % F;
        int ng = (o * NCI + c0 + cl) * F + f;
        int ngc = (l < LTOT) ? ng : 0;
#pragma unroll
        for (int kc = 0; kc < 8; ++kc) {
          int k = kc*4 + hi*2;
          v2f b;
          b.x = W3[ngc*32 + k];
          b.y = W3[ngc*32 + k + 1];
          acc = __builtin_amdgcn_wmma_f32_16x16x4_f32(false, aw[kc], false, b, (short)0, acc, false, false);
        }
      }
      // bias (clamped load; junk columns unused) + store R chunk
      {
        int l = lc*16 + lx;
        int o   = l / (L::CB * F);
        int clf = l % (L::CB * F);
        int cl = clf / F, f = clf % F;
        int ng = (o * NCI + c0 + cl) * F + f;
        float bb = b3[(l < LTOT) ? ng : 0];
#pragma unroll
        for (int rr = 0; rr < 8; ++rr)
          smem[L::Ro + (hi*8 + rr)*16 + lx] = acc[rr] + bb;
      }
      wave_lds_fence();
      // fold chunk into t[e,o,p]; lane halves split the p-range (no races)
      {
        const int e = lx;
        const int pBeg = hi ? ((MO + 1) / 2) : 0;
        const int pEnd = hi ? MO : ((MO + 1) / 2);
        for (int nn = 0; nn < 16; ++nn) {
          int ll = lc*16 + nn;
          if (ll >= LTOT) break;
          int oo = ll / (L::CB * F);
          int cf = ll % (L::CB * F);
          int cc = cf / F, ff = cf % F;
          float rv = smem[L::Ro + e*16 + nn];
          const float* yp = &smem[L::Yo + ((e*L::CB + cc)*MO)*F + ff];
          for (int p = pBeg; p < pEnd; ++p)
            smem[L::To + (e*NCO + oo)*MO + p] += rv * yp[p*F];
        }
      }
      wave_lds_fence();
    }
    wave_lds_fence();
  }

  // ---- write back ----------------------------------------------------------
  if (sdst != nullptr) {
    for (int i = lane; i < 16*NCO*MO; i += 32) {
      int e = i / (NCO*MO);
      atomicAdd(&out[(size_t)sdst[e0 + e]*(NCO*MO) + (i % (NCO*MO))], smem[L::To + i]);
    }
  } else {
    for (int i = lane; i < 16*NCO*MO; i += 32) {
      int e = i / (NCO*MO);
      size_t oi = (size_t)(e0 + e)*(NCO*MO) + (i % (NCO*MO));
      if (accumulate) out[oi] += smem[L::To + i];
      else            out[oi]  = smem[L::To + i];
    }
  }
}

// --------------------------- small kernels ---------------------------------
__global__ void fill_kernel(float* p, float v, long n) {
  long i = (long)blockIdx.x * blockDim.x + threadIdx.x;
  if (i < n) p[i] = v;
}

// y[n,o,m] = sum_c W[o,c] * x[n,c,m]
__global__ void node_linear_kernel(const float* __restrict__ W, const float* __restrict__ x,
                                   float* __restrict__ y, int N, int O, int C, int M)
{
  int i = blockIdx.x * blockDim.x + threadIdx.x;
  if (i >= N*O*M) return;
  int m = i % M; int t = i / M; int o = t % O; int n = t / O;
  float acc = 0.f;
  for (int c = 0; c < C; ++c) acc += W[o*C + c] * x[((size_t)n*C + c)*M + m];
  y[i] = acc;
}

// y[n,o,m] = W[o,:A]@xa[n,:,m] + W[o,A:A+B]@xb[n,:,m]   (skip-concat proj)
__global__ void proj_cat_kernel(const float* __restrict__ W, const float* __restrict__ xa,
                                const float* __restrict__ xb, float* __restrict__ y,
                                int N, int O, int A, int B, int M)
{
  int i = blockIdx.x * blockDim.x + threadIdx.x;
  if (i >= N*O*M) return;
  int m = i % M; int t = i / M; int o = t % O; int n = t / O;
  float acc = 0.f;
  for (int c = 0; c < A; ++c) acc += W[o*(A+B) + c]     * xa[((size_t)n*A + c)*M + m];
  for (int c = 0; c < B; ++c) acc += W[o*(A+B) + A + c] * xb[((size_t)n*B + c)*M + m];
  y[i] = acc;
}

__global__ void logits_kernel(const float* k0, const float* q0,
                              const float* k1, const float* q1,
                              const float* k2, const float* q2,
                              const float* k3, const float* q3,
                              const int* __restrict__ dst, float scale,
                              float* __restrict__ lg)
{
  int i = blockIdx.x * blockDim.x + threadIdx.x;
  if (i >= NE*4) return;
  int e = i >> 2, h = i & 3;
  int dn = dst[e];
  float acc = 0.f;
  if (k0) for (int j = 0; j < 1; ++j) acc += k0[(e*4+h)*1+j] * q0[(dn*4+h)*1+j];
  if (k1) for (int j = 0; j < 3; ++j) acc += k1[(e*4+h)*3+j] * q1[(dn*4+h)*3+j];
  if (k2) for (int j = 0; j < 5; ++j) acc += k2[(e*4+h)*5+j] * q2[(dn*4+h)*5+j];
  if (k3) for (int j = 0; j < 7; ++j) acc += k3[(e*4+h)*7+j] * q3[(dn*4+h)*7+j];
  lg[i] = acc * scale;
}

__device__ inline void atomicMaxFloat(float* a, float v) {
  int* ia = (int*)a;
  int cur = __float_as_int(*a);
  while (__int_as_float(cur) < v) {
    int prev = atomicCAS(ia, cur, __float_as_int(v));
    if (prev == cur) break;
    cur = prev;
  }
}

__global__ void segmax_kernel(const float* __restrict__ lg, const int* __restrict__ dst,
                              float* __restrict__ mx)
{
  int i = blockIdx.x * blockDim.x + threadIdx.x;
  if (i >= NE*4) return;
  atomicMaxFloat(&mx[dst[i >> 2]*4 + (i & 3)], lg[i]);
}

__global__ void expsum_kernel(const float* __restrict__ lg, const int* __restrict__ dst,
                              const float* __restrict__ mx, float* __restrict__ ex,
                              float* __restrict__ den)
{
  int i = blockIdx.x * blockDim.x + threadIdx.x;
  if (i >= NE*4) return;
  int idx = dst[i >> 2]*4 + (i & 3);
  float v = expf(lg[i] - mx[idx]);
  ex[i] = v;
  atomicAdd(&den[idx], v);
}

__global__ void alpha_kernel(const float* __restrict__ ex, const int* __restrict__ dst,
                             const float* __restrict__ den, float* __restrict__ al)
{
  int i = blockIdx.x * blockDim.x + threadIdx.x;
  if (i >= NE*4) return;
  al[i] = ex[i] / den[dst[i >> 2]*4 + (i & 3)];
}

__global__ void aggregate_kernel(const float* __restrict__ al, const float* __restrict__ v,
                                 const int* __restrict__ dst, float* __restrict__ o, int M)
{
  int i = blockIdx.x * blockDim.x + threadIdx.x;
  if (i >= NE*4*M) return;
  int e = i / (4*M); int rem = i % (4*M); int h = rem / M; int m = rem % M;
  atomicAdd(&o[((size_t)dst[e]*4 + h)*M + m], al[e*4 + h] * v[i]);
}

// in-place equivariant norm: x[c,m] *= relu(W@||x||+b)[c] / ||x[c]||
__global__ __launch_bounds__(32)
void gnorm_kernel(const float* __restrict__ W, const float* __restrict__ b,
                  float* __restrict__ x, int M)
{
  __shared__ float norms[16];
  __shared__ float scale[16];
  int n = blockIdx.x, lane = threadIdx.x;
  float* xn = x + (size_t)n*16*M;
  if (lane < 16) {
    float s = 1e-12f;
    for (int m = 0; m < M; ++m) { float v = xn[lane*M + m]; s += v*v; }
    norms[lane] = sqrtf(s);
  }
  __syncthreads();
  if (lane < 16) {
    float acc = b[lane];
    for (int c = 0; c < 16; ++c) acc += W[lane*16 + c] * norms[c];
    scale[lane] = acc > 0.f ? acc : 0.f;
  }
  __syncthreads();
  for (int i = lane; i < 16*M; i += 32) {
    int c = i / M;
    xn[i] = xn[i] * (scale[c] / norms[c]);
  }
}

__global__ void pool_kernel(const float* __restrict__ x, float* __restrict__ pooled, int OM) {
  int i = blockIdx.x * blockDim.x + threadIdx.x;
  if (i >= NN*OM) return;
  atomicAdd(&pooled[i % OM], x[i] * (1.f / NN));
}

__global__ void fc2_kernel(const float* __restrict__ W0, const float* __restrict__ p0,
                           const float* __restrict__ W1, const float* __restrict__ p1,
                           float* __restrict__ out)
{
  int i = threadIdx.x;
  if (i < 2) {
    float a = 0.f;
    for (int c = 0; c < 64; ++c) a += W0[i*64 + c] * p0[c];
    out[i] = a;
  } else if (i < 8) {
    int o = (i - 2) / 3, m = (i - 2) % 3;
    float a = 0.f;
    for (int c = 0; c < 64; ++c) a += W1[o*64 + c] * p1[c*3 + m];
    out[i] = a;
  }
}

// ------------------------------- host side ---------------------------------
struct Radial { const float *W1,*b1,*W2,*b2,*W3,*b3; };
static Radial take_radial(void* const* d_in, int& i) {
  Radial r;
  r.W1 = (const float*)d_in[i++]; r.b1 = (const float*)d_in[i++];
  r.W2 = (const float*)d_in[i++]; r.b2 = (const float*)d_in[i++];
  r.W3 = (const float*)d_in[i++]; r.b3 = (const float*)d_in[i++];
  return r;
}

static void launch_pconv(int di, int dd, int nci, int nco, const Radial& R,
                         const float* basis, const float* x,
                         const float* rp, const float* wp,
                         const int* src, const int* sdst,
                         float* out, int acc, hipStream_t stream)
{
#define PC(DI,DO,NCI_,NCO_)                                                    \
  if (di==DI && dd==DO && nci==NCI_ && nco==NCO_) {                            \
    constexpr int MI_ = 2*DI+1, MO_ = 2*DO+1, F_ = 2*((DI<DO)?DI:DO)+1;        \
    using LL = PL<NCI_,NCO_,MI_,MO_,F_>;                                       \
    pconv_kernel<NCI_,NCO_,MI_,MO_,F_>                                         \
      <<<dim3(NE/16), dim3(32), LL::TOTAL*sizeof(float), stream>>>(            \
        rp, wp, src, R.W1, R.b1, R.W2, R.b2, R.W3, R.b3, basis, x, sdst, out, acc); \
    return; }
  PC(0,0,1,4)  PC(0,1,1,4)  PC(0,2,1,4)  PC(0,3,1,4)
  PC(1,0,1,4)  PC(1,1,1,4)  PC(1,2,1,4)  PC(1,3,1,4)
  PC(0,0,16,4) PC(0,1,16,4) PC(0,2,16,4) PC(0,3,16,4)
  PC(1,0,16,4) PC(1,1,16,4) PC(1,2,16,4) PC(1,3,16,4)
  PC(2,0,16,4) PC(2,1,16,4) PC(2,2,16,4) PC(2,3,16,4)
  PC(3,0,16,4) PC(3,1,16,4) PC(3,2,16,4) PC(3,3,16,4)
  PC(0,0,16,64) PC(0,1,16,64) PC(1,0,16,64) PC(1,1,16,64)
  PC(2,0,16,64) PC(2,1,16,64) PC(3,0,16,64) PC(3,1,16,64)
#undef PC
}

extern "C" void kernel_launch(void* const* d_in, const int* in_sizes, int n_in,
                              void* d_out, int out_size, void* d_ws, size_t ws_size,
                              hipStream_t stream)
{
  (void)in_sizes; (void)n_in; (void)out_size; (void)ws_size;
  const float* mass     = (const float*)d_in[0];
  const float* velocity = (const float*)d_in[1];
  const float* wfeat    = (const float*)d_in[2];
  const float* rfeat    = (const float*)d_in[3];
  const int*   src      = (const int*)d_in[4];
  const int*   dst      = (const int*)d_in[5];
  int ix = 6;
  const float* basis[4][4];
  for (int di = 0; di < 4; ++di)
    for (int dd = 0; dd < 4; ++dd) basis[di][dd] = (const float*)d_in[ix++];

  struct LayerP {
    const float* q[4];
    Radial k[4][4], v[4][4];
    const float* proj[4];
    const float* nW[4]; const float* nb[4];
  } L0 = {}, L1 = {};
  // layer 0 (f_in degs {0,1})
  for (int d = 0; d < 2; ++d) L0.q[d] = (const float*)d_in[ix++];
  for (int di = 0; di < 2; ++di) for (int dd = 0; dd < 2; ++dd) L0.k[di][dd] = take_radial(d_in, ix);
  for (int di = 0; di < 2; ++di) for (int dd = 0; dd < 4; ++dd) L0.v[di][dd] = take_radial(d_in, ix);
  for (int d = 0; d < 4; ++d) L0.proj[d] = (const float*)d_in[ix++];
  for (int d = 0; d < 4; ++d) { L0.nW[d] = (const float*)d_in[ix++]; L0.nb[d] = (const float*)d_in[ix++]; }
  // layer 1 (f_in degs {0..3})
  for (int d = 0; d < 4; ++d) L1.q[d] = (const float*)d_in[ix++];
  for (int di = 0; di < 4; ++di) for (int dd = 0; dd < 4; ++dd) L1.k[di][dd] = take_radial(d_in, ix);
  for (int di = 0; di < 4; ++di) for (int dd = 0; dd < 4; ++dd) L1.v[di][dd] = take_radial(d_in, ix);
  for (int d = 0; d < 4; ++d) L1.proj[d] = (const float*)d_in[ix++];
  for (int d = 0; d < 4; ++d) { L1.nW[d] = (const float*)d_in[ix++]; L1.nb[d] = (const float*)d_in[ix++]; }
  // final conv + heads
  Radial convR[4][2];
  for (int di = 0; di < 4; ++di) for (int dd = 0; dd < 2; ++dd) convR[di][dd] = take_radial(d_in, ix);
  const float* convSelf[2]; for (int d = 0; d < 2; ++d) convSelf[d] = (const float*)d_in[ix++];
  const float* fc1p[2];     for (int d = 0; d < 2; ++d) fc1p[d]     = (const float*)d_in[ix++];
  const float* fc2p[2];     for (int d = 0; d < 2; ++d) fc2p[d]     = (const float*)d_in[ix++];

  // ----- workspace bump allocator (~150 MB) -----
  char* wp_ = (char*)d_ws;
  auto alloc = [&](size_t nf) {
    float* p = (float*)wp_;
    wp_ += ((nf * sizeof(float) + 255) & ~(size_t)255);
    return p;
  };
  float* h0buf = alloc((size_t)NN*16*16);
  float* h1buf = alloc((size_t)NN*16*16);
  float* kbuf  = alloc((size_t)NE*4*16);
  float* vbuf  = alloc((size_t)NE*4*16);
  float* qbuf  = alloc((size_t)NN*4*16);
  float* obuf  = alloc((size_t)NN*4*16);
  float* lg    = alloc((size_t)NE*4);
  float* exb   = alloc((size_t)NE*4);
  float* alb   = alloc((size_t)NE*4);
  float* mx    = alloc((size_t)NN*4);
  float* den   = alloc((size_t)NN*4);
  float* cn    = alloc((size_t)NN*64*4);
  float* f1b   = alloc((size_t)NN*64*4);
  float* pooled = alloc(64*4);

  const int mdeg[4] = {1,3,5,7};
  const int cumm[4] = {0,1,4,9};
  auto hsl = [&](float* base, int d) { return base + (size_t)NN*16*cumm[d]; };
  auto ks  = [&](int d) { return kbuf + (size_t)NE*4*cumm[d]; };
  auto vsl = [&](int d) { return vbuf + (size_t)NE*4*cumm[d]; };
  auto qsl = [&](int d) { return qbuf + (size_t)NN*4*cumm[d]; };
  auto osl = [&](int d) { return obuf + (size_t)NN*4*cumm[d]; };
  auto eb   = [](long n) { return dim3((unsigned)((n + 255) / 256)); };
  auto fill = [&](float* p, float vv, long n) {
    fill_kernel<<<eb(n), dim3(256), 0, stream>>>(p, vv, n);
  };

  // ============================ layer 0 ============================
  node_linear_kernel<<<eb((long)NN*4*1), dim3(256), 0, stream>>>(L0.q[0], mass,     qsl(0), NN, 4, 1, 1);
  node_linear_kernel<<<eb((long)NN*4*3), dim3(256), 0, stream>>>(L0.q[1], velocity, qsl(1), NN, 4, 1, 3);
  const float* x0[2] = { mass, velocity };
  for (int dd = 0; dd < 2; ++dd)
    for (int di = 0; di < 2; ++di)
      launch_pconv(di, dd, 1, 4, L0.k[di][dd], basis[di][dd], x0[di], rfeat, wfeat, src, nullptr, ks(dd), di > 0, stream);
  for (int dd = 0; dd < 4; ++dd)
    for (int di = 0; di < 2; ++di)
      launch_pconv(di, dd, 1, 4, L0.v[di][dd], basis[di][dd], x0[di], rfeat, wfeat, src, nullptr, vsl(dd), di > 0, stream);
  logits_kernel<<<eb((long)NE*4), dim3(256), 0, stream>>>(
      ks(0), qsl(0), ks(1), qsl(1), nullptr, nullptr, nullptr, nullptr, dst, 0.25f, lg);
  fill(mx, -1e30f, (long)NN*4); fill(den, 0.f, (long)NN*4);
  segmax_kernel<<<eb((long)NE*4), dim3(256), 0, stream>>>(lg, dst, mx);
  expsum_kernel<<<eb((long)NE*4), dim3(256), 0, stream>>>(lg, dst, mx, exb, den);
  alpha_kernel <<<eb((long)NE*4), dim3(256), 0, stream>>>(exb, dst, den, alb);
  for (int d = 0; d < 4; ++d) {
    fill(osl(d), 0.f, (long)NN*4*mdeg[d]);
    aggregate_kernel<<<eb((long)NE*4*mdeg[d]), dim3(256), 0, stream>>>(alb, vsl(d), dst, osl(d), mdeg[d]);
  }
  proj_cat_kernel<<<eb((long)NN*16*1), dim3(256), 0, stream>>>(L0.proj[0], osl(0), mass,     hsl(h0buf,0), NN, 16, 4, 1, 1);
  proj_cat_kernel<<<eb((long)NN*16*3), dim3(256), 0, stream>>>(L0.proj[1], osl(1), velocity, hsl(h0buf,1), NN, 16, 4, 1, 3);
  proj_cat_kernel<<<eb((long)NN*16*5), dim3(256), 0, stream>>>(L0.proj[2], osl(2), nullptr,  hsl(h0buf,2), NN, 16, 4, 0, 5);
  proj_cat_kernel<<<eb((long)NN*16*7), dim3(256), 0, stream>>>(L0.proj[3], osl(3), nullptr,  hsl(h0buf,3), NN, 16, 4, 0, 7);
  for (int d = 0; d < 4; ++d)
    gnorm_kernel<<<dim3(NN), dim3(32), 0, stream>>>(L0.nW[d], L0.nb[d], hsl(h0buf,d), mdeg[d]);

  // ============================ layer 1 ============================
  for (int d = 0; d < 4; ++d)
    node_linear_kernel<<<eb((long)NN*4*mdeg[d]), dim3(256), 0, stream>>>(L1.q[d], hsl(h0buf,d), qsl(d), NN, 4, 16, mdeg[d]);
  for (int dd = 0; dd < 4; ++dd)
    for (int di = 0; di < 4; ++di)
      launch_pconv(di, dd, 16, 4, L1.k[di][dd], basis[di][dd], hsl(h0buf,di), rfeat, wfeat, src, nullptr, ks(dd), di > 0, stream);
  for (int dd = 0; dd < 4; ++dd)
    for (int di = 0; di < 4; ++di)
      launch_pconv(di, dd, 16, 4, L1.v[di][dd], basis[di][dd], hsl(h0buf,di), rfeat, wfeat, src, nullptr, vsl(dd), di > 0, stream);
  logits_kernel<<<eb((long)NE*4), dim3(256), 0, stream>>>(
      ks(0), qsl(0), ks(1), qsl(1), ks(2), qsl(2), ks(3), qsl(3), dst, 0.125f, lg);
  fill(mx, -1e30f, (long)NN*4); fill(den, 0.f, (long)NN*4);
  segmax_kernel<<<eb((long)NE*4), dim3(256), 0, stream>>>(lg, dst, mx);
  expsum_kernel<<<eb((long)NE*4), dim3(256), 0, stream>>>(lg, dst, mx, exb, den);
  alpha_kernel <<<eb((long)NE*4), dim3(256), 0, stream>>>(exb, dst, den, alb);
  for (int d = 0; d < 4; ++d) {
    fill(osl(d), 0.f, (long)NN*4*mdeg[d]);
    aggregate_kernel<<<eb((long)NE*4*mdeg[d]), dim3(256), 0, stream>>>(alb, vsl(d), dst, osl(d), mdeg[d]);
  }
  for (int d = 0; d < 4; ++d)
    proj_cat_kernel<<<eb((long)NN*16*mdeg[d]), dim3(256), 0, stream>>>(
        L1.proj[d], osl(d), hsl(h0buf,d), hsl(h1buf,d), NN, 16, 4, 16, mdeg[d]);
  for (int d = 0; d < 4; ++d)
    gnorm_kernel<<<dim3(NN), dim3(32), 0, stream>>>(L1.nW[d], L1.nb[d], hsl(h1buf,d), mdeg[d]);

  // ===================== final conv + output heads =====================
  float* cn0 = cn; float* cn1 = cn + (size_t)NN*64*1;
  node_linear_kernel<<<eb((long)NN*64*1), dim3(256), 0, stream>>>(convSelf[0], hsl(h1buf,0), cn0, NN, 64, 16, 1);
  node_linear_kernel<<<eb((long)NN*64*3), dim3(256), 0, stream>>>(convSelf[1], hsl(h1buf,1), cn1, NN, 64, 16, 3);
  for (int dd = 0; dd < 2; ++dd)
    for (int di = 0; di < 4; ++di)
      launch_pconv(di, dd, 16, 64, convR[di][dd], basis[di][dd], hsl(h1buf,di),
                   rfeat, wfeat, src, dst, dd == 0 ? cn0 : cn1, 1, stream);
  float* f10 = f1b; float* f11 = f1b + (size_t)NN*64*1;
  node_linear_kernel<<<eb((long)NN*64*1), dim3(256), 0, stream>>>(fc1p[0], cn0, f10, NN, 64, 64, 1);
  node_linear_kernel<<<eb((long)NN*64*3), dim3(256), 0, stream>>>(fc1p[1], cn1, f11, NN, 64, 64, 3);
  float* p0 = pooled; float* p1 = pooled + 64;
  fill(pooled, 0.f, 64*4);
  pool_kernel<<<eb((long)NN*64),   dim3(256), 0, stream>>>(f10, p0, 64);
  pool_kernel<<<eb((long)NN*64*3), dim3(256), 0, stream>>>(f11, p1, 192);
  fc2_kernel<<<dim3(1), dim3(32), 0, stream>>>(fc2p[0], p0, fc2p[1], p1, (float*)d_out);
}